// GAT_51522427683147
// MI455X (gfx1250) — compile-verified
//
#include <hip/hip_runtime.h>
#include <hip/hip_bf16.h>

typedef __attribute__((ext_vector_type(2))) float v2f;
typedef __attribute__((ext_vector_type(8))) float v8f;

#define NEG_SLOPE 0.2f

// ---------- float <-> order-preserving unsigned encoding (for atomicMax) ----
__device__ __forceinline__ unsigned encf(float f) {
    unsigned u = __float_as_uint(f);
    return (u >> 31) ? ~u : (u | 0x80000000u);
}
__device__ __forceinline__ float decf(unsigned u) {
    return (u >> 31) ? __uint_as_float(u ^ 0x80000000u) : __uint_as_float(~u);
}
#define ENC_NEGINF 0x007FFFFFu   // encf(-inf)

// ---------------------------------------------------------------------------
// WMMA GEMM: Out[M x NCOLS] = A[M x 128] @ W[128 x NCOLS]   (fp32, exact-ish)
// Block = 256 threads (8 waves). Block tile: 32 rows x 64 cols.
// Each wave owns one 16x16 tile, K-loop of V_WMMA_F32_16X16X4_F32.
// ---------------------------------------------------------------------------
template<int NCOLS>
__global__ __launch_bounds__(256) void gemm_wmma(const float* __restrict__ A,
                                                 const float* __restrict__ W,
                                                 float* __restrict__ Out, int M) {
    constexpr int K  = 128;
    constexpr int KP = K + 4;                 // pad to kill LDS bank conflicts
    __shared__ float sA[32 * KP];             // [row][k]
    __shared__ float sW[64 * KP];             // [n][k]  (transposed)

    const int tid  = threadIdx.x;
    const int row0 = blockIdx.x * 32;
    const int col0 = blockIdx.y * 64;

    // Stage A tile (32x128), zero-pad out-of-range rows.
    for (int i = tid; i < 32 * K; i += 256) {
        int r = i >> 7, k = i & 127;
        int gr = row0 + r;
        sA[r * KP + k] = (gr < M) ? A[gr * K + k] : 0.0f;
    }
    // Stage W tile transposed: sW[n][k] = W[k][col0+n]
    for (int i = tid; i < K * 64; i += 256) {
        int k = i >> 6, n = i & 63;
        sW[n * KP + k] = W[k * NCOLS + col0 + n];
    }
    __syncthreads();

    const int wave   = tid >> 5;              // 0..7 -> 2 Mtiles x 4 Ntiles
    const int lane   = tid & 31;
    const int laneLo = lane & 15;
    const int hiHalf = lane >> 4;             // 0: K pair {0,1}, 1: {2,3}
    const int mt = wave >> 2;                 // 0..1
    const int nt = wave & 3;                  // 0..3

    const float* ap = &sA[(mt * 16 + laneLo) * KP + hiHalf * 2];
    const float* bp = &sW[(nt * 16 + laneLo) * KP + hiHalf * 2];

    v8f c = {};
#pragma unroll
    for (int kk = 0; kk < K; kk += 4) {
        v2f a = *(const v2f*)(ap + kk);       // A 16x4: lanes0-15 K={kk,kk+1}, lanes16-31 K={kk+2,kk+3}
        v2f b = *(const v2f*)(bp + kk);       // B 4x16 mirrored over N
        c = __builtin_amdgcn_wmma_f32_16x16x4_f32(false, a, false, b,
                                                  (short)0, c, false, false);
    }

    // C/D layout: VGPR r -> M = 8*hiHalf + r, N = laneLo
    const int gnBase = col0 + nt * 16 + laneLo;
    const int gmBase = row0 + mt * 16 + hiHalf * 8;
#pragma unroll
    for (int r = 0; r < 8; r++) {
        int gm = gmBase + r;
        if (gm < M) Out[gm * NCOLS + gnBase] = c[r];
    }
}

// ---------------------------------------------------------------------------
// Per-node attention logits: as[n,h] = dot(xl[n,h,:], att_src[h,:]), same for ad
// ---------------------------------------------------------------------------
__global__ void alpha_kernel(const float* __restrict__ xl,
                             const float* __restrict__ att_s,
                             const float* __restrict__ att_d,
                             float* __restrict__ as_, float* __restrict__ ad_,
                             int N, int H) {
    int i = blockIdx.x * blockDim.x + threadIdx.x;   // i = node*H + h
    if (i >= N * H) return;
    int h = i % H;
    const float* xp = xl + i * 64;
    const float* sp = att_s + h * 64;
    const float* dp = att_d + h * 64;
    float s = 0.f, d = 0.f;
#pragma unroll 4
    for (int c = 0; c < 64; c++) { float v = xp[c]; s += v * sp[c]; d += v * dp[c]; }
    as_[i] = s; ad_[i] = d;
}

__global__ void init_md(unsigned* __restrict__ m, float* __restrict__ denom, int n) {
    int i = blockIdx.x * blockDim.x + threadIdx.x;
    if (i < n) { m[i] = ENC_NEGINF; denom[i] = 0.f; }
}

template<int H>
__global__ void edge_max_kernel(const int* __restrict__ ei, int E, int N,
                                const float* __restrict__ as_, const float* __restrict__ ad_,
                                unsigned* __restrict__ m) {
    int e = blockIdx.x * blockDim.x + threadIdx.x;
    int TE = E + N;
    if (e >= TE) return;
    int s, d;
    if (e < E) { s = ei[e]; d = ei[E + e]; } else { s = d = e - E; }
#pragma unroll
    for (int h = 0; h < H; h++) {
        float v = as_[s * H + h] + ad_[d * H + h];
        v = v > 0.f ? v : NEG_SLOPE * v;
        atomicMax(&m[d * H + h], encf(v));
    }
}

__global__ void m_fix(unsigned* __restrict__ m, int n) {
    int i = blockIdx.x * blockDim.x + threadIdx.x;
    if (i >= n) return;
    float f = decf(m[i]);
    if (!isfinite(f)) f = 0.f;
    ((float*)m)[i] = f;
}

template<int H>
__global__ void edge_denom_kernel(const int* __restrict__ ei, int E, int N,
                                  const float* __restrict__ as_, const float* __restrict__ ad_,
                                  const float* __restrict__ mf, float* __restrict__ denom) {
    int e = blockIdx.x * blockDim.x + threadIdx.x;
    int TE = E + N;
    if (e >= TE) return;
    int s, d;
    if (e < E) { s = ei[e]; d = ei[E + e]; } else { s = d = e - E; }
#pragma unroll
    for (int h = 0; h < H; h++) {
        float v = as_[s * H + h] + ad_[d * H + h];
        v = v > 0.f ? v : NEG_SLOPE * v;
        atomicAdd(&denom[d * H + h], __expf(v - mf[d * H + h]));
    }
}

// One wave (32 lanes) per edge; each lane handles F/32 consecutive channels.
template<int H, int C>
__global__ __launch_bounds__(256) void edge_scatter(const int* __restrict__ ei, int E, int N,
                                                    const float* __restrict__ as_,
                                                    const float* __restrict__ ad_,
                                                    const float* __restrict__ mf,
                                                    const float* __restrict__ denom,
                                                    const float* __restrict__ xl,
                                                    float* __restrict__ acc) {
    constexpr int F   = H * C;
    constexpr int PER = F / 32;
    int wid  = (blockIdx.x * blockDim.x + threadIdx.x) >> 5;
    int lane = threadIdx.x & 31;
    int TE = E + N;
    if (wid >= TE) return;
    int s, d;
    if (wid < E) { s = ei[wid]; d = ei[E + wid]; } else { s = d = wid - E; }

    float alpha[H];
#pragma unroll
    for (int h = 0; h < H; h++) {
        float v = as_[s * H + h] + ad_[d * H + h];
        v = v > 0.f ? v : NEG_SLOPE * v;
        alpha[h] = __expf(v - mf[d * H + h]) / denom[d * H + h];
    }

    const float* xp = xl + s * F + lane * PER;
    float*       op = acc + d * F + lane * PER;
    float vals[PER];
#pragma unroll
    for (int j = 0; j < PER; j++) vals[j] = xp[j];
#pragma unroll
    for (int j = 0; j < PER; j++) {
        int h = (lane * PER + j) / C;
        atomicAdd(&op[j], vals[j] * alpha[h]);
    }
}

__global__ void finalize_kernel(float* __restrict__ acc, const float* __restrict__ bias,
                                int total, int F) {
    int i = blockIdx.x * blockDim.x + threadIdx.x;
    if (i >= total) return;
    float v = acc[i] + bias[i % F];
    acc[i] = v > 0.f ? v : 0.f;
}

// ---------------------------------------------------------------------------
extern "C" void kernel_launch(void* const* d_in, const int* in_sizes, int n_in,
                              void* d_out, int out_size, void* d_ws, size_t ws_size,
                              hipStream_t stream) {
    const float* x   = (const float*)d_in[0];
    const int*   ei  = (const int*)d_in[1];
    const float* W1  = (const float*)d_in[2];
    const float* as1 = (const float*)d_in[3];
    const float* ad1 = (const float*)d_in[4];
    const float* b1  = (const float*)d_in[5];
    const float* W2  = (const float*)d_in[6];
    const float* as2 = (const float*)d_in[7];
    const float* ad2 = (const float*)d_in[8];
    const float* b2  = (const float*)d_in[9];

    const int N = in_sizes[0] / 128;          // 50000
    const int E = in_sizes[1] / 2;            // 1600000
    const int TE = E + N;

    // workspace layout (floats). xl reused for layer-2 xl (N*64 <= N*128).
    float*    ws    = (float*)d_ws;
    float*    xl    = ws;                      // N*128
    float*    acc1  = ws + (size_t)N * 128;    // N*128 (becomes h1 in place)
    float*    asb   = acc1 + (size_t)N * 128;  // N*2
    float*    adb   = asb + (size_t)N * 2;     // N*2
    unsigned* mb    = (unsigned*)(adb + (size_t)N * 2); // N*2
    float*    denom = (float*)mb + (size_t)N * 2;       // N*2
    float*    out   = (float*)d_out;           // N*64 accumulator + result

    dim3 blk(256);
    const int mblocks = (N + 31) / 32;

    hipMemsetAsync(acc1, 0, (size_t)N * 128 * sizeof(float), stream);
    hipMemsetAsync(out,  0, (size_t)N * 64  * sizeof(float), stream);

    // -------- Layer 1 --------
    gemm_wmma<128><<<dim3(mblocks, 2), blk, 0, stream>>>(x, W1, xl, N);
    alpha_kernel<<<(N * 2 + 255) / 256, blk, 0, stream>>>(xl, as1, ad1, asb, adb, N, 2);
    init_md<<<(N * 2 + 255) / 256, blk, 0, stream>>>(mb, denom, N * 2);
    edge_max_kernel<2><<<(TE + 255) / 256, blk, 0, stream>>>(ei, E, N, asb, adb, mb);
    m_fix<<<(N * 2 + 255) / 256, blk, 0, stream>>>(mb, N * 2);
    edge_denom_kernel<2><<<(TE + 255) / 256, blk, 0, stream>>>(ei, E, N, asb, adb, (const float*)mb, denom);
    edge_scatter<2, 64><<<(TE + 7) / 8, blk, 0, stream>>>(ei, E, N, asb, adb, (const float*)mb, denom, xl, acc1);
    finalize_kernel<<<(N * 128 + 255) / 256, blk, 0, stream>>>(acc1, b1, N * 128, 128);

    // -------- Layer 2 --------
    gemm_wmma<64><<<dim3(mblocks, 1), blk, 0, stream>>>(acc1, W2, xl, N);
    alpha_kernel<<<(N + 255) / 256, blk, 0, stream>>>(xl, as2, ad2, asb, adb, N, 1);
    init_md<<<(N + 255) / 256, blk, 0, stream>>>(mb, denom, N);
    edge_max_kernel<1><<<(TE + 255) / 256, blk, 0, stream>>>(ei, E, N, asb, adb, mb);
    m_fix<<<(N + 255) / 256, blk, 0, stream>>>(mb, N);
    edge_denom_kernel<1><<<(TE + 255) / 256, blk, 0, stream>>>(ei, E, N, asb, adb, (const float*)mb, denom);
    edge_scatter<1, 64><<<(TE + 7) / 8, blk, 0, stream>>>(ei, E, N, asb, adb, (const float*)mb, denom, xl, out);
    finalize_kernel<<<(N * 64 + 255) / 256, blk, 0, stream>>>(out, b2, N * 64, 64);
}